// SparseEncoderSpatial_12214886990220
// MI455X (gfx1250) — compile-verified
//
#include <hip/hip_runtime.h>
#include <hip/hip_bf16.h>

typedef float v2f __attribute__((ext_vector_type(2)));
typedef float v8f __attribute__((ext_vector_type(8)));

#define SZ 256
#define NB 8
#define NE 64
#define NC 64
#define THRV 15.0f
#define THR2V 225.0f
#define FSTRIDE 66
#define FEATS_TOTAL (NB * SZ * SZ * FSTRIDE)  // 34,603,008

// ---------------------------------------------------------------------------
// Kernel 1: per-edge affine coefficients, laid out exactly as the WMMA
// B-matrix operand (4x16, f32): VGPR0 = {K0 | K2}, VGPR1 = {K1 | K3},
// lanes 0-15 = columns (edges) with (k0,k1), lanes 16-31 with (k2,k3).
// Stored as one float2 per lane -> coalesced b64 loads in the main kernel.
// ws layout: v2f ws[b][g][q][32]  (8*4*4*32*8B = 32 KB)
// ---------------------------------------------------------------------------
__global__ void edge_coef_kernel(const float* __restrict__ edges,
                                 float* __restrict__ wsB) {
  int b = blockIdx.x;   // 0..7
  int e = threadIdx.x;  // 0..63
  const float* ed = edges + (b * NE + e) * 4;
  float v0 = ed[0] * SZ, u0 = ed[1] * SZ, v1 = ed[2] * SZ, u1 = ed[3] * SZ;
  float dv = v1 - v0, du = u1 - u0;
  float len = sqrtf(dv * dv + du * du);
  float L = fmaxf(len, 1e-4f);
  float invL = 1.0f / L;
  float dirv = dv * invL, diru = du * invL;
  float nv = diru, nu = -dirv;  // normals = (dirs.u, -dirs.v)

  // A row = [v, u, 1, v*v+u*u]
  float k[4][4];
  // q0: normal_raw = n.(UV - p0)
  k[0][0] = nv; k[0][1] = nu; k[0][2] = -(v0 * nv + u0 * nu); k[0][3] = 0.0f;
  // q1: dir_d = d.(UV - p0) / L   (dirs already /L -> /L^2 on dvec)
  k[1][0] = dirv * invL; k[1][1] = diru * invL;
  k[1][2] = -(v0 * dirv + u0 * diru) * invL; k[1][3] = 0.0f;
  // q2: |UV - p0|^2
  k[2][0] = -2.0f * v0; k[2][1] = -2.0f * u0;
  k[2][2] = v0 * v0 + u0 * u0; k[2][3] = 1.0f;
  // q3: |UV - p1|^2
  k[3][0] = -2.0f * v1; k[3][1] = -2.0f * u1;
  k[3][2] = v1 * v1 + u1 * u1; k[3][3] = 1.0f;

  int g = e >> 4, n = e & 15;
  v2f* w2 = (v2f*)wsB;
  int base = ((b * 4 + g) * 4) * 32;
#pragma unroll
  for (int q = 0; q < 4; ++q) {
    w2[base + q * 32 + n]      = (v2f){k[q][0], k[q][1]};
    w2[base + q * 32 + 16 + n] = (v2f){k[q][2], k[q][3]};
  }
}

// ---------------------------------------------------------------------------
// Kernel 2: fused sample + edge-WMMA + masked feature write.
// Block = 256 thr = 8 waves; each wave owns a 16-pixel row chunk
// (v fixed, u = u0..u0+15) and processes all 8 batches for it.
// ---------------------------------------------------------------------------
__global__ __launch_bounds__(256) void sparse_encoder_kernel(
    const float* __restrict__ img, const float* __restrict__ wsB,
    float* __restrict__ out) {
  __shared__ float smem[8][1088];  // per wave: samp[16][65] + mask/p0/p1[16]
  const int lane = threadIdx.x & 31;
  const int w = threadIdx.x >> 5;
  const int chunk = blockIdx.x * 8 + w;  // 0..4095
  const int v = chunk >> 4;
  const int u0 = (chunk & 15) << 4;
  float* samp  = smem[w];
  float* smask = samp + 1040;
  float* sp0   = smask + 16;
  float* sp1   = sp0 + 16;

  const int p = lane & 15;
  const int half = lane >> 4;

  // ---- Phase 1: 2x2 zero-padded average (bilinear at j-0.5) into LDS ----
  {
    int row  = v - 1 + half;       // half 0 -> v-1, half 1 -> v (always <=255)
    int colA = u0 + p - 1;
    int colB = u0 + p;
    float fr = (row >= 0) ? 1.0f : 0.0f;
    float fa = (colA >= 0) ? fr : 0.0f;
    int rowc  = row  < 0 ? 0 : row;
    int colAc = colA < 0 ? 0 : colA;
    for (int c = 0; c < NC; ++c) {
      const float* bp = img + (c * SZ + rowc) * SZ;
      float s = fa * bp[colAc] + fr * bp[colB];
      s += __shfl_xor(s, 16);  // combine the two rows
      if (half == 0) samp[p * 65 + c] = 0.25f * s;
    }
  }

  // ---- Phase 2: A-matrix operand: [v, u, 1, v^2+u^2] per pixel ----
  float vf = (float)v;
  float uf = (float)(u0 + p);
  v2f A;
  A.x = half ? 1.0f : vf;                  // K=0 | K=2
  A.y = half ? (vf * vf + uf * uf) : uf;   // K=1 | K=3

  const v2f* wB = (const v2f*)wsB;
  const v8f zero8 = {0, 0, 0, 0, 0, 0, 0, 0};

  for (int b = 0; b < NB; ++b) {
    float acc_c[8], acc_0[8], acc_1[8];
#pragma unroll
    for (int r = 0; r < 8; ++r) { acc_c[r] = 0; acc_0[r] = 0; acc_1[r] = 0; }

#pragma unroll
    for (int g = 0; g < 4; ++g) {  // 16 edges per group
      int base = ((b * 4 + g) * 4) * 32 + lane;
      v2f b0 = wB[base + 0 * 32];
      v2f b1 = wB[base + 1 * 32];
      v2f b2 = wB[base + 2 * 32];
      v2f b3 = wB[base + 3 * 32];
      // D[m][n]: row m = pixel (r + 8*half), col n = edge (g*16 + lane&15)
      v8f dn = __builtin_amdgcn_wmma_f32_16x16x4_f32(false, A, false, b0, (short)0, zero8, false, false);
      v8f dd = __builtin_amdgcn_wmma_f32_16x16x4_f32(false, A, false, b1, (short)0, zero8, false, false);
      v8f d1 = __builtin_amdgcn_wmma_f32_16x16x4_f32(false, A, false, b2, (short)0, zero8, false, false);
      v8f d2 = __builtin_amdgcn_wmma_f32_16x16x4_f32(false, A, false, b3, (short)0, zero8, false, false);
#pragma unroll
      for (int r = 0; r < 8; ++r) {
        float nd = fabsf(dn[r]);
        float dird = dd[r];
        bool m = (nd <= THRV && dird <= 1.0f && dird >= 0.0f) ||
                 (d1[r] <= THR2V) || (d2[r] <= THR2V);
        float mf = m ? 1.0f : 0.0f;
        acc_c[r] += mf;
        acc_0[r] += mf * fmaxf(dird, 1.0f - dird);
        acc_1[r] += mf * (1.0f - nd * (1.0f / THRV));
      }
    }

    // reduce over the 16 edge-columns (lanes) within each half
#pragma unroll
    for (int d = 1; d < 16; d <<= 1) {
#pragma unroll
      for (int r = 0; r < 8; ++r) {
        acc_c[r] += __shfl_xor(acc_c[r], d);
        acc_0[r] += __shfl_xor(acc_0[r], d);
        acc_1[r] += __shfl_xor(acc_1[r], d);
      }
    }
    if ((lane & 15) == 0) {  // lane 0 -> pixels 0..7, lane 16 -> 8..15
#pragma unroll
      for (int r = 0; r < 8; ++r) {
        int pp = r + half * 8;
        float cnt = acc_c[r];
        float inv = 1.0f / fmaxf(cnt, 1e-4f);
        smask[pp] = (cnt > 0.0f) ? 1.0f : 0.0f;
        sp0[pp] = acc_0[r] * inv;
        sp1[pp] = acc_1[r] * inv;
      }
    }
    __syncthreads();

    // ---- masked feature write: 16 px * 66 = 1056 contiguous floats ----
    long fb = ((long)(b * SZ + v) * SZ + u0) * FSTRIDE;
    for (int t = lane; t < 16 * FSTRIDE; t += 32) {
      int pp = t / FSTRIDE;
      int cmp = t - pp * FSTRIDE;
      float val = (cmp < 64) ? samp[pp * 65 + cmp]
                             : ((cmp == 64) ? sp0[pp] : sp1[pp]);
      out[fb + t] = val * smask[pp];
    }
    if (half == 0) {
      out[(long)FEATS_TOTAL + (long)b * SZ * SZ + v * SZ + u0 + p] = smask[p];
    }
    __syncthreads();
  }
}

extern "C" void kernel_launch(void* const* d_in, const int* in_sizes, int n_in,
                              void* d_out, int out_size, void* d_ws,
                              size_t ws_size, hipStream_t stream) {
  (void)in_sizes; (void)n_in; (void)out_size; (void)ws_size;
  const float* img   = (const float*)d_in[0];  // [1,64,256,256]
  const float* edges = (const float*)d_in[1];  // [8,64,4]
  float* out = (float*)d_out;                  // feats [8,256,256,66] ++ masks [8,256,256]
  float* wsB = (float*)d_ws;                   // 32 KB of B-matrix coeffs

  edge_coef_kernel<<<dim3(NB), dim3(NE), 0, stream>>>(edges, wsB);
  sparse_encoder_kernel<<<dim3(512), dim3(256), 0, stream>>>(img, wsB, out);
}